// OctreeConv_82609400971431
// MI455X (gfx1250) — compile-verified
//
#include <hip/hip_runtime.h>
#include <stdint.h>

#define NKD    27
#define CIN    32
#define COUT   32
#define CHUNKS (CIN / 4)      // 8 K-chunks of 4 (f32 WMMA K=4)
#define TPW    4              // 16-node tiles per wave (amortizes B fragments 4x)
#define WAVES_PER_BLOCK 8

typedef __attribute__((ext_vector_type(2))) float v2f;
typedef __attribute__((ext_vector_type(8))) float v8f;

__global__ __launch_bounds__(256, 4)
void OctreeConv_wmma_f32(const float* __restrict__ data,
                         const long long* __restrict__ neigh,
                         const float* __restrict__ w,
                         float* __restrict__ out,
                         int n_nodes)
{
    // B fragments in WMMA lane order: 27*8*2*32 float2 = 110592 B
    extern __shared__ v2f wlds[];

    const int tid = threadIdx.x;

    // One-time stage: swizzle weights (k, ci, co) into per-lane B-fragment order.
    // B tile 4x16 (f32): lane l -> N = l%16, K = vgpr + 2*(l/16)
    for (int idx = tid; idx < NKD * CHUNKS * 2 * 32; idx += blockDim.x) {
        const int ls  = idx & 31;
        const int h   = (idx >> 5) & 1;
        const int c   = (idx >> 6) & 7;
        const int j   = idx >> 9;
        const int kA  = 4 * c + 2 * (ls >> 4);
        const int col = 16 * h + (ls & 15);
        v2f v;
        v.x = w[(j * CIN + kA + 0) * COUT + col];
        v.y = w[(j * CIN + kA + 1) * COUT + col];
        wlds[idx] = v;
    }
    __syncthreads();

    const int lane = tid & 31;
    const int wave = tid >> 5;
    const int m    = lane & 15;   // A: M row, B/D: N col
    const int kh   = lane >> 4;   // half select: A K-offset 2*kh, D row-offset 8*kh

    const int tile0 = (blockIdx.x * WAVES_PER_BLOCK + wave) * TPW;
    if (tile0 * 16 >= n_nodes) return;   // wave-uniform: EXEC stays all-1s

    // Fixed per-lane index base; per-tile/per-j displacements become immediates.
    const char* nbase = (const char*)(neigh + (long long)(tile0 * 16 + m) * NKD);
    #pragma unroll
    for (int t = 0; t < TPW; ++t)
        __builtin_prefetch(nbase + t * (16 * NKD * 8), 0, 0);

    v8f acc[TPW][2];
    #pragma unroll
    for (int t = 0; t < TPW; ++t) {
        acc[t][0] = (v8f){};
        acc[t][1] = (v8f){};
    }

    const char* dbase = (const char*)data;

    #pragma unroll 3
    for (int j = 0; j < NKD; ++j) {
        // 32-bit gather offsets: r*128B + lane-half*8B (saddr + voffset + imm form).
        uint32_t aoff[TPW];
        #pragma unroll
        for (int t = 0; t < TPW; ++t) {
            const long long r = *(const long long*)(nbase + t * (16 * NKD * 8) + j * 8);
            aoff[t] = (uint32_t)r * (CIN * 4) + (uint32_t)(kh * 8);
        }

        const v2f* bj = wlds + j * (CHUNKS * 2 * 32) + lane;
        #pragma unroll
        for (int c = 0; c < CHUNKS; ++c) {
            // One conflict-free ds_load_b64 per B fragment, reused by 4 tiles (8 WMMAs).
            const v2f b0 = bj[(c * 2 + 0) * 32];
            const v2f b1 = bj[(c * 2 + 1) * 32];

            #pragma unroll
            for (int t = 0; t < TPW; ++t) {
                // A tile 16x4 (f32): lane l -> M = l%16, K = vgpr + 2*(l/16)
                const v2f a = *(const v2f*)(dbase + aoff[t] + c * 16);

                acc[t][0] = __builtin_amdgcn_wmma_f32_16x16x4_f32(
                    false, a, false, b0, (short)0, acc[t][0], false, false);
                acc[t][1] = __builtin_amdgcn_wmma_f32_16x16x4_f32(
                    false, a, false, b1, (short)0, acc[t][1], false, false);
            }
        }
    }

    // D layout 16x16 (f32): vgpr g, lane l -> row M = g + 8*(l/16), col N = l%16
    // All stores share one lane voffset; t/g displacements fold into immediates.
    float* obase = out + (size_t)tile0 * (16 * COUT);
    const uint32_t sofs = (uint32_t)(kh * 8 * COUT + m);   // floats
    #pragma unroll
    for (int t = 0; t < TPW; ++t) {
        #pragma unroll
        for (int g = 0; g < 8; ++g) {
            obase[sofs + t * (16 * COUT) + g * COUT]      = acc[t][0][g];
            obase[sofs + t * (16 * COUT) + g * COUT + 16] = acc[t][1][g];
        }
    }
}

extern "C" void kernel_launch(void* const* d_in, const int* in_sizes, int n_in,
                              void* d_out, int out_size, void* d_ws, size_t ws_size,
                              hipStream_t stream) {
    (void)n_in; (void)out_size; (void)d_ws; (void)ws_size;

    const float*     data  = (const float*)d_in[0];      // (N, 32) f32
    const long long* neigh = (const long long*)d_in[1];  // (N, 27) i64
    const float*     w     = (const float*)d_in[2];      // (27, 32, 32) f32
    float*           out   = (float*)d_out;              // (N, 32) f32

    const int n     = in_sizes[0] / CIN;
    const int tiles = (n + 15) / 16;
    const int waves = (tiles + TPW - 1) / TPW;
    const int blocks = (waves + WAVES_PER_BLOCK - 1) / WAVES_PER_BLOCK;
    const size_t shmem = (size_t)NKD * CHUNKS * 2 * 32 * sizeof(v2f);  // 110592 B

    // >64KB dynamic LDS (WGP pool is 320KB; 2 workgroups fit). Host-side call,
    // graph-capture safe and idempotent.
    hipFuncSetAttribute((const void*)OctreeConv_wmma_f32,
                        hipFuncAttributeMaxDynamicSharedMemorySize, (int)shmem);

    hipLaunchKernelGGL(OctreeConv_wmma_f32, dim3(blocks), dim3(256), shmem, stream,
                       data, neigh, w, out, n);
}